// GNNEncoder_75668733821211
// MI455X (gfx1250) — compile-verified
//
#include <hip/hip_runtime.h>
#include <hip/hip_bf16.h>

#define NN 100000
#define NE 1600000
#define NG 512
#define IN_DIM 64
#define HID 128

typedef __attribute__((ext_vector_type(2))) float v2f;
typedef __attribute__((ext_vector_type(8))) float v8f;

// ---------------- utility kernels ----------------
__global__ void gcn_set_val(float* __restrict__ p, float v, int n) {
    int i = blockIdx.x * 256 + threadIdx.x;
    if (i < n) p[i] = v;
}

__global__ void gcn_deg_add(const int* __restrict__ dst, float* __restrict__ deg) {
    int e = blockIdx.x * 256 + threadIdx.x;
    if (e < NE) atomicAdd(&deg[dst[e]], 1.0f);
}

__global__ void gcn_rsqrt_ip(float* __restrict__ d, int n) {
    int i = blockIdx.x * 256 + threadIdx.x;
    if (i < n) {
        float v = d[i];
        d[i] = v > 0.0f ? __frsqrt_rn(v) : 0.0f;
    }
}

// ---------------- WMMA GEMM: out[r][c] = dinv[r] * sum_k A[r][k]*W[k][c] ----------------
// A: [NN x K] row-major, W: [K x 128] row-major, out: [NN x 128].
// Block = 256 threads = 8 waves; wave w computes 16x16 tile at (blockIdx.x, w).
// A tile (16xK) and W chunk (64x128) staged in LDS (<= 40KB).
template <int K>
__global__ __launch_bounds__(256) void gcn_gemm_scaled(
    const float* __restrict__ A, const float* __restrict__ W,
    const float* __restrict__ dinv, float* __restrict__ out) {
    __shared__ float sA[16 * K];
    __shared__ float sW[64 * 128];
    const int tid = threadIdx.x;
    const float* Ab = A + (long)blockIdx.x * 16 * K;
    for (int i = tid; i < 16 * K; i += 256) sA[i] = Ab[i];

    const int wave = tid >> 5;
    const int lane = tid & 31;
    const int half = lane >> 4;   // which 16-lane half
    const int l16  = lane & 15;
    const int col  = wave * 16 + l16;

    v8f c = {};
    for (int k0 = 0; k0 < K; k0 += 64) {
        __syncthreads();  // sA ready (first iter) / sW reads done (later iters)
        for (int i = tid; i < 64 * 128; i += 256) sW[i] = W[(long)k0 * 128 + i];
        __syncthreads();
#pragma unroll
        for (int kk = 0; kk < 64; kk += 4) {
            // A 16x4 f32 layout: lanes 0-15 hold K={kk,kk+1}, lanes 16-31 K={kk+2,kk+3}
            const int ka = k0 + kk + 2 * half;
            v2f a, b;
            a.x = sA[l16 * K + ka];
            a.y = sA[l16 * K + ka + 1];
            // B 4x16 f32 layout mirrors A in K, row-striped across lanes in N
            const int kb = kk + 2 * half;
            b.x = sW[kb * 128 + col];
            b.y = sW[(kb + 1) * 128 + col];
            c = __builtin_amdgcn_wmma_f32_16x16x4_f32(
                false, a, false, b, (short)0, c, false, false);
        }
    }
    // D 16x16 f32: VGPR r -> M = r + 8*half, N = l16 (within tile)
#pragma unroll
    for (int r = 0; r < 8; ++r) {
        long row = (long)blockIdx.x * 16 + r + 8 * half;
        out[row * 128 + col] = c[r] * dinv[row];
    }
}

// ---------------- edge scatter: acc[dst] += g[src], 128 feats/edge ----------------
__global__ __launch_bounds__(256) void gcn_scatter_add(
    const int* __restrict__ src, const int* __restrict__ dst,
    const float* __restrict__ g, float* __restrict__ acc) {
    int e = blockIdx.x * 2 + (threadIdx.x >> 7);
    int f = threadIdx.x & 127;
    int s = src[e];
    int d = dst[e];
    atomicAdd(&acc[(long)d * 128 + f], g[(long)s * 128 + f]);
}

// ---------------- out = relu(dinv[row]*acc + bias[f]) ----------------
__global__ void gcn_bias_relu_scale(const float* __restrict__ acc,
                                    const float* __restrict__ dinv,
                                    const float* __restrict__ bias,
                                    float* __restrict__ out, int n) {
    int i = blockIdx.x * 256 + threadIdx.x;
    if (i < n) {
        int row = i >> 7;
        int f = i & 127;
        float v = dinv[row] * acc[i] + bias[f];
        out[i] = v > 0.0f ? v : 0.0f;
    }
}

// ---------------- mean pool ----------------
__global__ void gcn_pool_accum(const float* __restrict__ h, const int* __restrict__ batch,
                               float* __restrict__ sums, float* __restrict__ cnt) {
    int node = blockIdx.x;
    int f = threadIdx.x;  // 128
    int g = batch[node];
    atomicAdd(&sums[(long)g * 128 + f], h[(long)node * 128 + f]);
    if (f == 0) atomicAdd(&cnt[g], 1.0f);
}

__global__ void gcn_pool_div(const float* __restrict__ sums, const float* __restrict__ cnt,
                             float* __restrict__ out) {
    int i = blockIdx.x * 256 + threadIdx.x;  // NG*128
    float c = cnt[i >> 7];
    out[i] = sums[i] / (c > 1.0f ? c : 1.0f);
}

extern "C" void kernel_launch(void* const* d_in, const int* in_sizes, int n_in,
                              void* d_out, int out_size, void* d_ws, size_t ws_size,
                              hipStream_t stream) {
    const float* x   = (const float*)d_in[0];
    const int*   ei  = (const int*)d_in[1];
    const int*   bat = (const int*)d_in[2];
    const float* W1  = (const float*)d_in[3];
    const float* b1  = (const float*)d_in[4];
    const float* W2  = (const float*)d_in[5];
    const float* b2  = (const float*)d_in[6];
    float*       out = (float*)d_out;

    const int* src = ei;
    const int* dst = ei + NE;

    // workspace layout (floats)
    float* ws   = (float*)d_ws;
    float* dinv = ws;                                // NN (padded to 100352)
    float* bufA = ws + 100352;                       // NN*128
    float* bufB = bufA + (long)NN * 128;             // NN*128
    float* sums = bufB + (long)NN * 128;             // NG*128
    float* cnt  = sums + (long)NG * 128;             // NG
    const size_t tileBytes = (size_t)NN * 128 * sizeof(float);

    const int nThread = 256;
    const int gN   = (NN + 255) / 256;
    const int gE   = (NE + 255) / 256;
    const int gNF  = (NN * 128 + 255) / 256;

    // degrees with self-loop, then dinv = rsqrt(deg)
    gcn_set_val<<<gN, nThread, 0, stream>>>(dinv, 1.0f, NN);
    gcn_deg_add<<<gE, nThread, 0, stream>>>(dst, dinv);
    gcn_rsqrt_ip<<<gN, nThread, 0, stream>>>(dinv, NN);

    // ---- layer 1 ----
    gcn_gemm_scaled<IN_DIM><<<NN / 16, nThread, 0, stream>>>(x, W1, dinv, bufA);   // g1
    hipMemcpyAsync(bufB, bufA, tileBytes, hipMemcpyDeviceToDevice, stream);        // acc1 = g1 (self loop)
    gcn_scatter_add<<<NE / 2, nThread, 0, stream>>>(src, dst, bufA, bufB);
    gcn_bias_relu_scale<<<gNF, nThread, 0, stream>>>(bufB, dinv, b1, bufA, NN * 128);  // h1

    // ---- layer 2 ----
    gcn_gemm_scaled<HID><<<NN / 16, nThread, 0, stream>>>(bufA, W2, dinv, bufB);   // g2
    hipMemcpyAsync(bufA, bufB, tileBytes, hipMemcpyDeviceToDevice, stream);        // acc2 = g2
    gcn_scatter_add<<<NE / 2, nThread, 0, stream>>>(src, dst, bufB, bufA);
    gcn_bias_relu_scale<<<gNF, nThread, 0, stream>>>(bufA, dinv, b2, bufB, NN * 128);  // h2

    // ---- global mean pool ----
    gcn_set_val<<<(NG * 128 + NG + 255) / 256, nThread, 0, stream>>>(sums, 0.0f, NG * 128 + NG);
    gcn_pool_accum<<<NN, 128, 0, stream>>>(bufB, bat, sums, cnt);
    gcn_pool_div<<<(NG * 128) / 256, nThread, 0, stream>>>(sums, cnt, out);
}